// BsplineLoss_41910290874688
// MI455X (gfx1250) — compile-verified
//
#include <hip/hip_runtime.h>

#define B_    16
#define P_    4096
#define M_    1024
#define NCTRL 4
#define T_    16
#define NPRIM 64

typedef float v2f __attribute__((ext_vector_type(2)));
typedef float v8f __attribute__((ext_vector_type(8)));

// workspace layout in float units
#define OFF_CURVES 0                         // B_*M_*4 floats (x,y,z,|b|^2)
#define OFF_SKEL   (B_ * M_ * 4)             // B_*P_*4 floats (-2x,-2y,-2z,|a|^2)
#define OFF_COLMIN (OFF_SKEL + B_ * P_ * 4)  // B_*M_ uints (d2 col mins, +inf init)
#define OFF_ROWSUM (OFF_COLMIN + B_ * M_)    // B_ floats

__global__ void __launch_bounds__(256) prep_kernel(const float* __restrict__ skel,
                                                   const float* __restrict__ prm,
                                                   const float* __restrict__ basis,
                                                   float* __restrict__ ws) {
  const int idx = blockIdx.x * blockDim.x + threadIdx.x;  // 0 .. B_*P_-1
  float4* skel4 = (float4*)(ws + OFF_SKEL);
  float4* cur4  = (float4*)(ws + OFF_CURVES);
  unsigned* colmin = (unsigned*)(ws + OFF_COLMIN);
  float* rowsum = ws + OFF_ROWSUM;

  if (idx < B_ * P_) {
    float x = skel[idx * 3 + 0];
    float y = skel[idx * 3 + 1];
    float z = skel[idx * 3 + 2];
    // A pre-scaled by -2 so WMMA directly produces -2*a.b
    skel4[idx] = make_float4(-2.f * x, -2.f * y, -2.f * z, x * x + y * y + z * z);
  }
  if (idx < B_ * M_) {
    const int b = idx / M_, m = idx % M_;
    const int k = m / T_, t = m % T_;
    const float* cp = prm + (b * NPRIM + k) * (NCTRL * 3);
    const float* bs = basis + t * NCTRL;
    float x = 0.f, y = 0.f, z = 0.f;
#pragma unroll
    for (int n = 0; n < NCTRL; ++n) {
      const float w = bs[n];
      x += w * cp[n * 3 + 0];
      y += w * cp[n * 3 + 1];
      z += w * cp[n * 3 + 2];
    }
    cur4[idx] = make_float4(x, y, z, x * x + y * y + z * z);
    colmin[idx] = 0x7F800000u;  // +inf
  }
  if (idx < B_) rowsum[idx] = 0.f;
}

// grid: (P_/128, B_), block 256 = 8 waves; each wave owns a 16-row strip and
// sweeps all M in 16-col tiles via V_WMMA_F32_16X16X4_F32. Curve data is staged
// block-wide into LDS with async global->LDS copies.
__global__ void __launch_bounds__(256) chamfer_kernel(float* __restrict__ ws) {
  __shared__ float4   lcur[M_];   // 16 KB: curve points (x,y,z,|b|^2) for this batch
  __shared__ unsigned lcol[M_];   //  4 KB: per-block column d2 mins (u32 bit-order)

  const int b    = blockIdx.y;
  const int tid  = threadIdx.x;
  const int wave = tid >> 5;
  const int lane = tid & 31;
  const int half = lane >> 4;   // wave32 lane halves
  const int l16  = lane & 15;

  const float4* skel4 = (const float4*)(ws + OFF_SKEL) + (size_t)b * P_;
  const float4* cur4  = (const float4*)(ws + OFF_CURVES) + (size_t)b * M_;
  unsigned* colmin = (unsigned*)(ws + OFF_COLMIN) + (size_t)b * M_;
  float* rowsum = ws + OFF_ROWSUM + b;

  // ---- stage curves into LDS via CDNA5 async global->LDS copy ----
  {
    const unsigned lds_base = (unsigned)(size_t)(&lcur[0]);   // LDS byte offset
    const unsigned long long gbase = (unsigned long long)(size_t)cur4;
    for (int i = tid; i < M_; i += 256) {
      unsigned ldsa = lds_base + (unsigned)i * 16u;
      unsigned voff = (unsigned)i * 16u;
      asm volatile("global_load_async_to_lds_b128 %0, %1, %2"
                   :: "v"(ldsa), "v"(voff), "s"(gbase)
                   : "memory");
    }
  }
  for (int i = tid; i < M_; i += 256) lcol[i] = 0x7F800000u;
  asm volatile("s_wait_asynccnt 0" ::: "memory");
  __syncthreads();

  const int r0 = blockIdx.x * 128 + wave * 16;

  // A fragment (16x4 f32): lanes 0-15 hold (K0,K1)=(-2x,-2y), lanes 16-31 (K2,K3)=(-2z,0)
  const float4 sa = skel4[r0 + l16];
  v2f afrag;
  afrag.x = half ? sa.z : sa.x;
  afrag.y = half ? 0.f  : sa.y;

  // |a|^2 for the 8 D rows this lane holds: rows r0 + j + 8*half (broadcast loads)
  float aav[8];
#pragma unroll
  for (int j = 0; j < 8; ++j) aav[j] = skel4[r0 + j + 8 * half].w;

  float rmin[8];
#pragma unroll
  for (int j = 0; j < 8; ++j) rmin[j] = __builtin_inff();

  for (int m0 = 0; m0 < M_; m0 += 16) {
    const float4 cb = lcur[m0 + l16];  // ds_load_b128, block-shared
    // B fragment (4x16 f32): lanes 0-15 hold rows (K0,K1)=(x,y), lanes 16-31 (K2,K3)=(z,0)
    v2f bfrag;
    bfrag.x = half ? cb.z : cb.x;
    bfrag.y = half ? 0.f  : cb.y;
    const float bbv = cb.w;

    // fold |a|^2 + |b|^2 into the accumulator: D = (-2A)B + C  ==  d^2 tile
    v8f cin;
#pragma unroll
    for (int j = 0; j < 8; ++j) cin[j] = aav[j] + bbv;

    v8f d2 = __builtin_amdgcn_wmma_f32_16x16x4_f32(
        false, afrag, false, bfrag, (short)0, cin, false, false);

    float cm = d2[0];
#pragma unroll
    for (int j = 0; j < 8; ++j) {
      rmin[j] = fminf(rmin[j], d2[j]);
      if (j) cm = fminf(cm, d2[j]);
    }
    // branch-free column min: both halves issue non-returning ds_min_u32
    // (clamped >= 0 so u32 ordering == float ordering)
    cm = fmaxf(cm, 0.f);
    atomicMin(&lcol[m0 + l16], __float_as_uint(cm));
  }

  // row-min across the 16 lanes of each half (each lane covered cols ≡ l16 mod 16)
#pragma unroll
  for (int j = 0; j < 8; ++j) {
    float v = rmin[j];
    v = fminf(v, __shfl_xor(v, 1, 32));
    v = fminf(v, __shfl_xor(v, 2, 32));
    v = fminf(v, __shfl_xor(v, 4, 32));
    v = fminf(v, __shfl_xor(v, 8, 32));
    rmin[j] = v;
  }
  if (l16 == 0) {
    float s = 0.f;
#pragma unroll
    for (int j = 0; j < 8; ++j) s += sqrtf(fmaxf(rmin[j], 0.f));
    atomicAdd(rowsum, s);  // rows r0+j+8*half, 8 per half-leader
  }

  __syncthreads();
  for (int i = tid; i < M_; i += 256) atomicMin(&colmin[i], lcol[i]);
}

__global__ void __launch_bounds__(256) finalize_kernel(const float* __restrict__ ws,
                                                       float* __restrict__ out) {
  __shared__ float red[256];
  const int b = blockIdx.x;
  const unsigned* colmin = (const unsigned*)(ws + OFF_COLMIN) + (size_t)b * M_;

  float s = 0.f;
  for (int i = threadIdx.x; i < M_; i += 256)
    s += sqrtf(fmaxf(__uint_as_float(colmin[i]), 0.f));
  red[threadIdx.x] = s;
  __syncthreads();
  for (int off = 128; off > 0; off >>= 1) {
    if (threadIdx.x < off) red[threadIdx.x] += red[threadIdx.x + off];
    __syncthreads();
  }
  if (threadIdx.x == 0) {
    const float rows = ws[OFF_ROWSUM + b];
    out[b] = rows / (float)P_ + red[0] / (float)M_;  // * SCALE (== 1.0)
  }
}

extern "C" void kernel_launch(void* const* d_in, const int* in_sizes, int n_in,
                              void* d_out, int out_size, void* d_ws, size_t ws_size,
                              hipStream_t stream) {
  const float* skel  = (const float*)d_in[0];  // (16, 4096, 3)
  const float* prm   = (const float*)d_in[1];  // (16, 64, 12)
  const float* basis = (const float*)d_in[2];  // (16, 4)
  float* ws  = (float*)d_ws;
  float* out = (float*)d_out;                  // (16,)

  prep_kernel<<<dim3((B_ * P_) / 256), 256, 0, stream>>>(skel, prm, basis, ws);
  chamfer_kernel<<<dim3(P_ / 128, B_), 256, 0, stream>>>(ws);
  finalize_kernel<<<dim3(B_), 256, 0, stream>>>(ws, out);
}